// MyModule_70162585747929
// MI455X (gfx1250) — compile-verified
//
#include <hip/hip_runtime.h>
#include <math.h>

typedef __attribute__((ext_vector_type(2))) float v2f;
typedef __attribute__((ext_vector_type(4))) float v4f;
typedef __attribute__((ext_vector_type(8))) float v8f;

#define B_      8
#define T_      2048
#define H_      1024
#define DEPTH_  3
#define MTOT    (B_ * T_)      // 16384
#define GBLK    64             // persistent blocks in scan
#define NSL     (H_ / GBLK)    // 16 output features per scan block
#define WPITCH  1036           // LDS row pitch for Wh slice: 16B-aligned rows,
                               // 12-bank row stride -> conflict-free frag reads

// ---------------------------------------------------------------------------
// Phase 1: Pre[m][n] = sum_k In[m][k] * W[n][k] + bias[n]
// Written to outp with row stride ldo (the out[:,:,layer,:] slice).
// Block = 256 threads = 8 waves in a 4(M) x 2(N) grid; block tile 64x32;
// K staged through LDS in chunks of 64. f32 WMMA 16x16x4, dual accumulators.
// ---------------------------------------------------------------------------
__global__ __launch_bounds__(256)
void rnn_pre_gemm(const float* __restrict__ in, long ldi,
                  const float* __restrict__ w,
                  const float* __restrict__ bias,
                  float* __restrict__ outp, long ldo)
{
    __shared__ float sA[64][68];   // In chunk  [64 rows m][64 k], even pitch
    __shared__ float sB[32][68];   // W  chunk  [32 rows n][64 k]

    const int tid  = threadIdx.x;
    const int lane = tid & 31;
    const int wv   = tid >> 5;          // wave 0..7
    const int wm   = wv & 3;            // wave M index 0..3
    const int wn   = wv >> 2;           // wave N index 0..1
    const int m0   = blockIdx.x * 64;
    const int n0   = blockIdx.y * 32;

    const int lrow = lane & 15;
    const int koff = (lane >> 4) * 2;   // lanes 16..31 take K+2,K+3

    v8f acc0 = {}, acc1 = {};
    for (int k0 = 0; k0 < H_; k0 += 64) {
        // stage In 64x64 (every thread: 16 floats = 4x b128)
        {
            const int r = tid >> 2;
            const int c = (tid & 3) * 16;
            const float* g = in + (long)(m0 + r) * ldi + k0 + c;
            if (k0 + 64 < H_) __builtin_prefetch(g + 64, 0, 1);
            v4f* dst = (v4f*)&sA[r][c];
            #pragma unroll
            for (int i = 0; i < 4; ++i) dst[i] = *(const v4f*)(g + 4 * i);
        }
        // stage W 32x64 (threads 0..127: 16 floats each, 16B aligned)
        if (tid < 128) {
            const int r = tid >> 2;
            const int c = (tid & 3) * 16;
            const float* g = w + (long)(n0 + r) * H_ + k0 + c;
            v4f* dst = (v4f*)&sB[r][c];
            #pragma unroll
            for (int i = 0; i < 4; ++i) dst[i] = *(const v4f*)(g + 4 * i);
        }
        __syncthreads();

        #pragma unroll
        for (int kk = 0; kk < 64; kk += 8) {
            v2f a0 = *(const v2f*)&sA[wm * 16 + lrow][kk + koff];
            v2f b0 = *(const v2f*)&sB[wn * 16 + lrow][kk + koff];
            v2f a1 = *(const v2f*)&sA[wm * 16 + lrow][kk + 4 + koff];
            v2f b1 = *(const v2f*)&sB[wn * 16 + lrow][kk + 4 + koff];
            acc0 = __builtin_amdgcn_wmma_f32_16x16x4_f32(
                       false, a0, false, b0, (short)0, acc0, false, false);
            acc1 = __builtin_amdgcn_wmma_f32_16x16x4_f32(
                       false, a1, false, b1, (short)0, acc1, false, false);
        }
        __syncthreads();
    }

    // C layout: VGPR s, lanes0-15 -> (M=s, N=lane); lanes16-31 -> (M=s+8, N=lane-16)
    const int cn    = n0 + wn * 16 + (lane & 15);
    const int mbase = m0 + wm * 16 + ((lane >> 4) << 3);
    const float bv  = bias[cn];
    #pragma unroll
    for (int s = 0; s < 8; ++s)
        outp[(long)(mbase + s) * ldo + cn] = acc0[s] + acc1[s] + bv;
}

// ---------------------------------------------------------------------------
// Phase 2 init: zero h double-buffer (incl. pad rows 8..15) and step barriers.
// ---------------------------------------------------------------------------
__global__ void rnn_scan_init(float* __restrict__ hbuf, int* __restrict__ bar)
{
    const int i = blockIdx.x * blockDim.x + threadIdx.x;
    if (i < 2 * 16 * H_) hbuf[i] = 0.0f;
    if (i < T_)          bar[i]  = 0;
}

// ---------------------------------------------------------------------------
// Phase 2: persistent recurrent scan for one layer.
//   h_t = tanh(pre_t + h_{t-1} @ Wh^T)
// io points at out[:,:,k,:] (row stride DEPTH_*H_): holds pre in, h out.
// 64 blocks x 256 thr; block owns 16 output features, whose Wh rows live in
// LDS for all T steps; 8 waves split K=1024; device barrier per timestep.
// ---------------------------------------------------------------------------
__global__ __launch_bounds__(256)
void rnn_scan(const float* __restrict__ wh,   // [H][H] layer slice, row n, col k
              float* __restrict__ io,         // out + k*H_, row stride DEPTH_*H_
              float* __restrict__ hbuf,       // [2][16][H_]
              int* __restrict__ bar)          // [T_] arrival counters (zeroed)
{
    __shared__ float swh[16][WPITCH];         // this block's Wh slice (66.3 KB)
    __shared__ float red[8 * 256];            // per-wave partial C tiles (8 KB)

    const int tid  = threadIdx.x;
    const int lane = tid & 31;
    const int wv   = tid >> 5;
    const int n0   = blockIdx.x * NSL;
    const int lrow = lane & 15;
    const int koff = (lane >> 4) * 2;
    const long ldo = (long)DEPTH_ * H_;

    // One-time preload of Wh slice: thread -> row tid>>4, 64 cols (16x b128).
    {
        const int r  = tid >> 4;              // 0..15
        const int c0 = (tid & 15) * 64;
        const float* g = wh + (long)(n0 + r) * H_ + c0;
        #pragma unroll
        for (int i = 0; i < 16; ++i)
            *(v4f*)&swh[r][c0 + 4 * i] = *(const v4f*)(g + 4 * i);
    }
    __syncthreads();

    for (int t = 0; t < T_; ++t) {
        const float* hin  = hbuf + (t & 1) * (16 * H_);
        float*       hout = hbuf + ((t + 1) & 1) * (16 * H_);

        // K-partial 16x16 tile: this wave covers K in [kbase, kbase+128).
        // Dual accumulators break the WMMA->WMMA RAW chain; kbase is
        // wave-uniform so LDS offsets stay immediate.
        v8f acc0 = {}, acc1 = {};
        const int kbase = wv * 128;
        const float*       arow = hin + lrow * H_ + kbase + koff;
        const float* const brow = &swh[lrow][kbase + koff];
        #pragma unroll 4
        for (int kk = 0; kk < 128; kk += 8) {
            v2f a0 = *(const v2f*)(arow + kk);        // h rows (pad>=8 zero)
            v2f b0 = *(const v2f*)(brow + kk);        // Wh slice from LDS
            v2f a1 = *(const v2f*)(arow + kk + 4);
            v2f b1 = *(const v2f*)(brow + kk + 4);
            acc0 = __builtin_amdgcn_wmma_f32_16x16x4_f32(
                       false, a0, false, b0, (short)0, acc0, false, false);
            acc1 = __builtin_amdgcn_wmma_f32_16x16x4_f32(
                       false, a1, false, b1, (short)0, acc1, false, false);
        }

        // cross-wave reduction of the 8 partial tiles
        #pragma unroll
        for (int s = 0; s < 8; ++s)
            red[wv * 256 + s * 32 + lane] = acc0[s] + acc1[s];
        __syncthreads();

        float c = 0.0f;
        #pragma unroll
        for (int w = 0; w < 8; ++w) c += red[w * 256 + tid];

        // decode C position for this thread
        const int s  = tid >> 5;
        const int pl = tid & 31;
        const int m  = s + ((pl >> 4) << 3);        // 0..15 (batch, padded)
        const int n  = n0 + (pl & 15);

        float hval = 0.0f;
        if (m < B_) {
            const long row = (long)m * T_ + t;
            const float pre = io[row * ldo + n];
            hval = tanhf(c + pre);
            io[row * ldo + n] = hval;               // unmasked layer output
        }
        hout[m * H_ + n] = hval;                    // pad rows stay 0

        // device-wide step barrier
        __threadfence();
        __syncthreads();
        if (tid == 0) {
            __hip_atomic_fetch_add(&bar[t], 1, __ATOMIC_RELEASE,
                                   __HIP_MEMORY_SCOPE_AGENT);
            while (__hip_atomic_load(&bar[t], __ATOMIC_ACQUIRE,
                                     __HIP_MEMORY_SCOPE_AGENT) < GBLK) { }
        }
        __syncthreads();
        __threadfence();
    }
}

// ---------------------------------------------------------------------------
// Phase 3: zero positions t >= seq_len[b] (reference masks only at the end).
// ---------------------------------------------------------------------------
__global__ void rnn_mask(float* __restrict__ out, const int* __restrict__ seq)
{
    const long i = (long)blockIdx.x * blockDim.x + threadIdx.x;
    const long total = (long)B_ * T_ * DEPTH_ * H_;
    if (i >= total) return;
    const int b = (int)(i / ((long)T_ * DEPTH_ * H_));
    const int t = (int)((i / ((long)DEPTH_ * H_)) % T_);
    if (t >= seq[b]) out[i] = 0.0f;
}

// ---------------------------------------------------------------------------
extern "C" void kernel_launch(void* const* d_in, const int* in_sizes, int n_in,
                              void* d_out, int out_size, void* d_ws, size_t ws_size,
                              hipStream_t stream)
{
    const float* x     = (const float*)d_in[0];  // [B,T,H]
    const int*   seq   = (const int*)  d_in[1];  // [B]
    const float* W_ih  = (const float*)d_in[2];  // [DEPTH,H,H]
    const float* W_hh  = (const float*)d_in[3];  // [DEPTH,H,H]
    const float* bias  = (const float*)d_in[4];  // [DEPTH,H]
    float* out = (float*)d_out;                  // [B,T,DEPTH,H]

    float* hbuf = (float*)d_ws;                                  // 2*16*H floats
    int*   bar  = (int*)((char*)d_ws + 2 * 16 * H_ * sizeof(float));

    const long ldo = (long)DEPTH_ * H_;
    for (int k = 0; k < DEPTH_; ++k) {
        const float* in  = (k == 0) ? x : (out + (long)(k - 1) * H_);
        const long   ldi = (k == 0) ? (long)H_ : ldo;

        dim3 gg(MTOT / 64, H_ / 32);
        rnn_pre_gemm<<<gg, 256, 0, stream>>>(in, ldi,
                                             W_ih + (long)k * H_ * H_,
                                             bias + (long)k * H_,
                                             out + (long)k * H_, ldo);

        rnn_scan_init<<<(2 * 16 * H_ + 255) / 256, 256, 0, stream>>>(hbuf, bar);

        rnn_scan<<<GBLK, 256, 0, stream>>>(W_hh + (long)k * H_ * H_,
                                           out + (long)k * H_, hbuf, bar);
    }

    const long total = (long)B_ * T_ * DEPTH_ * H_;
    rnn_mask<<<(int)((total + 255) / 256), 256, 0, stream>>>(out, seq);
}